// RASF_34437047779690
// MI455X (gfx1250) — compile-verified
//
#include <hip/hip_runtime.h>
#include <math.h>

// Problem constants (from reference setup_inputs)
#define BATCH 16
#define NPTS  2048
#define KNN   32
#define CH    16
#define FD    32
#define FH    32
#define FW    32
#define CH_STRIDE (FD * FH * FW)   // 32768 floats per channel

typedef float v2f __attribute__((ext_vector_type(2)));
typedef float v8f __attribute__((ext_vector_type(8)));

// ---------------------------------------------------------------------------
// Kernel 0: zero the zoom accumulator (ws is poisoned with 0xAA by harness)
// ---------------------------------------------------------------------------
__global__ void RASF_init_kernel(unsigned* zoom_bits) { *zoom_bits = 0u; }

// ---------------------------------------------------------------------------
// Kernel 1: fused pairwise-distance (fp32 WMMA 16x16x4) + top-32 selection.
// One wave (32 lanes) owns 32 query points of one batch. For each tile of 16
// candidates we issue two V_WMMA_F32_16X16X4_F32 ops (A pre-scaled by -2 so
// the accumulator holds -2*inner exactly in fp32), assemble the 32x16
// distance tile in LDS, and each lane updates its private top-32 list
// (replace-worst + rescan) kept in LDS.
// ---------------------------------------------------------------------------
__global__ __launch_bounds__(32) void RASF_knn_kernel(
    const float* __restrict__ pts, int* __restrict__ idx_out)
{
    const int lane   = threadIdx.x;                  // 0..31
    const int chunks = NPTS / 32;                    // 64
    const int b      = blockIdx.x / chunks;
    const int base   = (blockIdx.x % chunks) * 32;   // first query of block
    const float* bp  = pts + (size_t)b * NPTS * 3;

    __shared__ float qx[32], qy[32], qz[32], qxx[32];
    __shared__ float cxx[16];
    __shared__ float dist[32][17];
    __shared__ float bd[32][33];
    __shared__ int   bi[32][33];

    // Load this block's 32 query points, cache coords + squared norms.
    {
        const float* q = bp + (size_t)(base + lane) * 3;
        float x = q[0], y = q[1], z = q[2];
        qx[lane] = x; qy[lane] = y; qz[lane] = z;
        qxx[lane] = x * x + y * y + z * z;
    }
    for (int j = 0; j < KNN; ++j) { bd[lane][j] = 3.4e38f; bi[lane][j] = 0; }
    __syncthreads();

    // A operands per ISA 32-bit 16x4 layout:
    //   lanes 0-15 : VGPR0=K0, VGPR1=K1 ; lanes 16-31 : VGPR0=K2, VGPR1=K3.
    // K = (x, y, z, 0), pre-scaled by -2 so WMMA returns -2*<q,c>.
    v2f a_lo, a_hi;
    {
        const int r = lane & 15;
        if (lane < 16) {
            a_lo.x = -2.0f * qx[r];      a_lo.y = -2.0f * qy[r];
            a_hi.x = -2.0f * qx[16 + r]; a_hi.y = -2.0f * qy[16 + r];
        } else {
            a_lo.x = -2.0f * qz[r];      a_lo.y = 0.0f;
            a_hi.x = -2.0f * qz[16 + r]; a_hi.y = 0.0f;
        }
    }

    float worst = 3.4e38f;
    int   worst_pos = 0;

    for (int tile = 0; tile < NPTS / 16; ++tile) {
        const int col = lane & 15;
        const int n   = tile * 16 + col;
        const float* c = bp + (size_t)n * 3;
        float x = c[0], y = c[1], z = c[2];
        if (lane < 16) cxx[col] = x * x + y * y + z * z;

        // B operand (4x16): rows K striped across VGPRs, N across lanes.
        v2f bmat;
        bmat.x = (lane < 16) ? x : z;
        bmat.y = (lane < 16) ? y : 0.0f;
        __syncthreads();

        v8f acc_lo = {};
        v8f acc_hi = {};
        acc_lo = __builtin_amdgcn_wmma_f32_16x16x4_f32(
            false, a_lo, false, bmat, (short)0, acc_lo, false, false);
        acc_hi = __builtin_amdgcn_wmma_f32_16x16x4_f32(
            false, a_hi, false, bmat, (short)0, acc_hi, false, false);

        // C layout: VGPR r holds row (r or r+8), lane%16 is the column.
        const int rbase = (lane < 16) ? 0 : 8;
#pragma unroll
        for (int r = 0; r < 8; ++r) {
            const int m = rbase + r;
            dist[m][col]      = acc_lo[r] + qxx[m]      + cxx[col];
            dist[16 + m][col] = acc_hi[r] + qxx[16 + m] + cxx[col];
        }
        __syncthreads();

        // Each lane scans its own query row; replace current worst on hit.
        for (int c2 = 0; c2 < 16; ++c2) {
            const float d = dist[lane][c2];
            if (d < worst) {
                bd[lane][worst_pos] = d;
                bi[lane][worst_pos] = tile * 16 + c2;
                worst = bd[lane][0]; worst_pos = 0;
#pragma unroll
                for (int j = 1; j < KNN; ++j) {
                    const float v = bd[lane][j];
                    if (v > worst) { worst = v; worst_pos = j; }
                }
            }
        }
        __syncthreads();
    }

    int* o = idx_out + ((size_t)b * NPTS + base + lane) * KNN;
    for (int j = 0; j < KNN; ++j) o[j] = bi[lane][j];
}

// ---------------------------------------------------------------------------
// Kernel 2: global zoom = max |rel| (== max(|min rel|, |max rel|)).
// Block-level reduction, one uint-bit-pattern atomicMax per block.
// ---------------------------------------------------------------------------
__global__ __launch_bounds__(256) void RASF_zoom_kernel(
    const float* __restrict__ pts, const int* __restrict__ idx,
    unsigned* __restrict__ zoom_bits)
{
    const int t = blockIdx.x * blockDim.x + threadIdx.x;   // 0 .. B*P-1
    const int b = t / NPTS, p = t % NPTS;
    const float* bp = pts + (size_t)b * NPTS * 3;
    const float px = bp[p * 3 + 0], py = bp[p * 3 + 1], pz = bp[p * 3 + 2];
    const int* ids = idx + (size_t)t * KNN;

    float m = 0.0f;
    for (int j = 0; j < KNN; ++j) {
        const int n = ids[j];
        const float rx = bp[n * 3 + 0] - px;
        const float ry = bp[n * 3 + 1] - py;
        const float rz = bp[n * 3 + 2] - pz;
        m = fmaxf(m, fmaxf(fabsf(rx), fmaxf(fabsf(ry), fabsf(rz))));
    }

    __shared__ float red[256];
    red[threadIdx.x] = m;
    __syncthreads();
    for (int s = 128; s > 0; s >>= 1) {
        if (threadIdx.x < s)
            red[threadIdx.x] = fmaxf(red[threadIdx.x], red[threadIdx.x + s]);
        __syncthreads();
    }
    if (threadIdx.x == 0)
        atomicMax(zoom_bits, __float_as_uint(red[0]));  // valid for floats >= 0
}

// ---------------------------------------------------------------------------
// Kernel 3: trilinear border sample of the shared 16x32^3 field + max over K.
// One thread per (b, p): 32 neighbors x 8 corners x 16 channels; the 2 MB
// field is L2-resident so the gathers are latency-, not HBM-, bound.
// ---------------------------------------------------------------------------
__global__ __launch_bounds__(256) void RASF_sample_kernel(
    const float* __restrict__ pts, const float* __restrict__ field,
    const int* __restrict__ idx, const unsigned* __restrict__ zoom_bits,
    float* __restrict__ out)
{
    const int t = blockIdx.x * blockDim.x + threadIdx.x;
    if (t >= BATCH * NPTS) return;
    const int b = t / NPTS, p = t % NPTS;

    const float inv_zoom = 1.0f / __uint_as_float(*zoom_bits);
    const float* bp = pts + (size_t)b * NPTS * 3;
    const float px = bp[p * 3 + 0], py = bp[p * 3 + 1], pz = bp[p * 3 + 2];
    const int* ids = idx + (size_t)t * KNN;

    float cm[CH];
#pragma unroll
    for (int c = 0; c < CH; ++c) cm[c] = -3.4e38f;

    for (int j = 0; j < KNN; ++j) {
        const int n = ids[j];
        const float x = (bp[n * 3 + 0] - px) * inv_zoom;   // -> W
        const float y = (bp[n * 3 + 1] - py) * inv_zoom;   // -> H
        const float z = (bp[n * 3 + 2] - pz) * inv_zoom;   // -> D

        // align_corners=False unnormalize: ((c+1)*size - 1) * 0.5, size = 32
        const float ix = (x + 1.0f) * 16.0f - 0.5f;
        const float iy = (y + 1.0f) * 16.0f - 0.5f;
        const float iz = (z + 1.0f) * 16.0f - 0.5f;
        const float fx0 = floorf(ix), fy0 = floorf(iy), fz0 = floorf(iz);
        const float fx = ix - fx0, fy = iy - fy0, fz = iz - fz0;

        const int x0 = min(max((int)fx0, 0), FW - 1);
        const int x1 = min(max((int)fx0 + 1, 0), FW - 1);
        const int y0 = min(max((int)fy0, 0), FH - 1);
        const int y1 = min(max((int)fy0 + 1, 0), FH - 1);
        const int z0 = min(max((int)fz0, 0), FD - 1);
        const int z1 = min(max((int)fz0 + 1, 0), FD - 1);

        const float wx0 = 1.0f - fx, wy0 = 1.0f - fy, wz0 = 1.0f - fz;
        const int o00 = z0 * (FH * FW) + y0 * FW;
        const int o01 = z0 * (FH * FW) + y1 * FW;
        const int o10 = z1 * (FH * FW) + y0 * FW;
        const int o11 = z1 * (FH * FW) + y1 * FW;

        const float w[8] = { wz0 * wy0 * wx0, wz0 * wy0 * fx,
                             wz0 * fy  * wx0, wz0 * fy  * fx,
                             fz  * wy0 * wx0, fz  * wy0 * fx,
                             fz  * fy  * wx0, fz  * fy  * fx };
        const int off[8] = { o00 + x0, o00 + x1, o01 + x0, o01 + x1,
                             o10 + x0, o10 + x1, o11 + x0, o11 + x1 };

#pragma unroll
        for (int c = 0; c < CH; ++c) {
            const float* fc = field + (size_t)c * CH_STRIDE;
            float v = 0.0f;
#pragma unroll
            for (int q = 0; q < 8; ++q) v += w[q] * fc[off[q]];
            cm[c] = fmaxf(cm[c], v);
        }
    }

    float* ob = out + (size_t)b * CH * NPTS;   // (B, C, P)
#pragma unroll
    for (int c = 0; c < CH; ++c) ob[(size_t)c * NPTS + p] = cm[c];
}

// ---------------------------------------------------------------------------
// Host-side launcher
// ---------------------------------------------------------------------------
extern "C" void kernel_launch(void* const* d_in, const int* in_sizes, int n_in,
                              void* d_out, int out_size, void* d_ws, size_t ws_size,
                              hipStream_t stream) {
    const float* pts   = (const float*)d_in[0];  // (16, 2048, 3)
    const float* field = (const float*)d_in[1];  // (1, 16, 32, 32, 32)
    float* out = (float*)d_out;                  // (16, 16, 2048)

    // Workspace layout: [ idx: B*P*K ints = 4 MB ][ zoom bits: 1 uint ]
    int*      idx  = (int*)d_ws;
    unsigned* zoom = (unsigned*)((char*)d_ws +
                                 (size_t)BATCH * NPTS * KNN * sizeof(int));

    RASF_init_kernel<<<1, 1, 0, stream>>>(zoom);
    RASF_knn_kernel<<<BATCH * (NPTS / 32), 32, 0, stream>>>(pts, idx);
    RASF_zoom_kernel<<<(BATCH * NPTS) / 256, 256, 0, stream>>>(pts, idx, zoom);
    RASF_sample_kernel<<<(BATCH * NPTS + 255) / 256, 256, 0, stream>>>(
        pts, field, idx, zoom, out);
}